// FairnessLoss_77558519431603
// MI455X (gfx1250) — compile-verified
//
#include <hip/hip_runtime.h>
#include <math.h>

// ---------------------------------------------------------------------------
// FairnessLoss on MI455X (gfx1250, wave32)
//
// Roofline: logits = 262144 x 1000 f32 = 1.05 GB read once -> ~45us @ 23.3TB/s.
// Compute (~1 GFLOP of exp/max/add) is negligible -> bandwidth-bound design:
//   Stage 1: one wave per row, full row in registers (8 x float4 per lane),
//            one-pass logsumexp, per-group LDS accumulation, per-block partials.
//   Stage 2: deterministic segment reduction of partials via chained
//            V_WMMA_F32_16X16X4_F32 (A = ones 16x4, D = A*B + D), then scalar
//            epilogue for base loss + 0.1*sqrt(var(group_means) + 1e-8).
// ---------------------------------------------------------------------------

#define NROWS   262144
#define NCOLS   1000
#define NGROUPS 8

#define S1_BLOCKS   1024
#define S1_THREADS  256                      // 8 waves of 32
#define WAVES_PER_BLOCK (S1_THREADS / 32)
#define TOTAL_WAVES (S1_BLOCKS * WAVES_PER_BLOCK)   // 8192
#define ROWS_PER_WAVE (NROWS / TOTAL_WAVES)         // 32
#define PARTIAL_W (2 * NGROUPS)              // 8 sums + 8 counts = 16 floats/block

typedef __attribute__((ext_vector_type(2))) float v2f;
typedef __attribute__((ext_vector_type(8))) float v8f;

// ---------------------------------------------------------------------------
// Stage 1: per-row logsumexp + per-group accumulation -> per-block partials
// ---------------------------------------------------------------------------
__global__ __launch_bounds__(S1_THREADS)
void fairness_stage1(const float* __restrict__ logits,
                     const int*   __restrict__ targets,
                     const int*   __restrict__ group_ids,
                     float*       __restrict__ partials)
{
    __shared__ float acc[WAVES_PER_BLOCK][PARTIAL_W];

    const int tid  = threadIdx.x;
    const int lane = tid & 31;
    const int wv   = tid >> 5;

    if (lane == 0) {
#pragma unroll
        for (int i = 0; i < PARTIAL_W; ++i) acc[wv][i] = 0.0f;
    }

    const int gwave = blockIdx.x * WAVES_PER_BLOCK + wv;

    for (int it = 0; it < ROWS_PER_WAVE; ++it) {
        const int r = gwave + it * TOTAL_WAVES;
        const float* __restrict__ row = logits + (size_t)r * NCOLS;

        // Prefetch this wave's next row into L2 (-> global_prefetch_b8).
        // 32 lanes * 128B stride covers the 4000B row.
        if (it + 1 < ROWS_PER_WAVE) {
            __builtin_prefetch(row + (size_t)TOTAL_WAVES * NCOLS + lane * 32, 0, 0);
        }

        // Entire row into registers: 8 x float4 per lane (1024 slots, pad -inf).
        float4 v[8];
#pragma unroll
        for (int j = 0; j < 7; ++j) {
            v[j] = *(const float4*)(row + j * 128 + lane * 4);
        }
        if (lane < 26) {            // 896 + 26*4 = 1000 exactly
            v[7] = *(const float4*)(row + 896 + lane * 4);
        } else {
            const float ninf = -__builtin_inff();
            v[7] = make_float4(ninf, ninf, ninf, ninf);
        }

        // Lane-local max, then wave32 max.
        float m = -__builtin_inff();
#pragma unroll
        for (int j = 0; j < 8; ++j) {
            m = fmaxf(m, fmaxf(fmaxf(v[j].x, v[j].y), fmaxf(v[j].z, v[j].w)));
        }
#pragma unroll
        for (int off = 16; off > 0; off >>= 1) {
            m = fmaxf(m, __shfl_xor(m, off, 32));
        }

        // Sum of exp(x - m); padded -inf lanes contribute exp(-inf) = 0.
        float s = 0.0f;
#pragma unroll
        for (int j = 0; j < 8; ++j) {
            s += __expf(v[j].x - m) + __expf(v[j].y - m) +
                 __expf(v[j].z - m) + __expf(v[j].w - m);
        }
#pragma unroll
        for (int off = 16; off > 0; off >>= 1) {
            s += __shfl_xor(s, off, 32);
        }

        if (lane == 0) {
            const int   t    = targets[r];
            const float tl   = row[t];                 // L0-resident (just streamed)
            const float loss = __logf(s) + m - tl;     // logsumexp - logit[target]
            const int   g    = group_ids[r] & (NGROUPS - 1);
            acc[wv][g]           += loss;              // single-writer LDS slots
            acc[wv][NGROUPS + g] += 1.0f;
        }
    }

    __syncthreads();

    if (tid < PARTIAL_W) {
        float s = 0.0f;
#pragma unroll
        for (int w = 0; w < WAVES_PER_BLOCK; ++w) s += acc[w][tid];
        partials[blockIdx.x * PARTIAL_W + tid] = s;
    }
}

// ---------------------------------------------------------------------------
// Stage 2: reduce 1024 x 16 partials with V_WMMA_F32_16X16X4_F32.
// A = ones(16x4)  =>  D = A*B + D accumulates column sums of each 4x16 chunk.
// (All-ones A makes the result independent of B's K-slot lane layout.)
// ---------------------------------------------------------------------------
__global__ __launch_bounds__(128)
void fairness_stage2(const float* __restrict__ partials, float* __restrict__ out)
{
    __shared__ float cs[4][PARTIAL_W];

    const int tid  = threadIdx.x;
    const int lane = tid & 31;
    const int wv   = tid >> 5;

    v2f a; a.x = 1.0f; a.y = 1.0f;     // A-matrix: all ones (layout-invariant)
    v8f d = {};                        // accumulator C/D

    const int col  = lane & 15;
    const int half = lane >> 4;        // lanes 0-15 take rows p,p+1; 16-31 rows p+2,p+3
    const int rows_per_wave = S1_BLOCKS / 4;     // 256 partial rows per wave
    const int base = wv * rows_per_wave;

#pragma unroll 4
    for (int p = 0; p < rows_per_wave; p += 4) {
        const int r0 = base + p + half * 2;
        v2f b;
        b.x = partials[(r0    ) * PARTIAL_W + col];
        b.y = partials[(r0 + 1) * PARTIAL_W + col];
        // 8 args: (neg_a, A, neg_b, B, c_mod, C, reuse_a, reuse_b)
        d = __builtin_amdgcn_wmma_f32_16x16x4_f32(
                false, a, false, b, (short)0, d, false, false);
    }

    // D row M=0: VGPR0, lanes 0-15 hold N=0..15 -> the 16 column totals.
    if (lane < 16) cs[wv][lane] = d[0];
    __syncthreads();

    if (tid == 0) {
        float sums[NGROUPS], cnts[NGROUPS];
        float total = 0.0f;
#pragma unroll
        for (int g = 0; g < NGROUPS; ++g) {
            sums[g] = cs[0][g] + cs[1][g] + cs[2][g] + cs[3][g];
            cnts[g] = cs[0][NGROUPS + g] + cs[1][NGROUPS + g] +
                      cs[2][NGROUPS + g] + cs[3][NGROUPS + g];
            total  += sums[g];
        }
        const float base_loss = total / (float)NROWS;

        float gm[NGROUPS], mean_of_means = 0.0f;
#pragma unroll
        for (int g = 0; g < NGROUPS; ++g) {
            gm[g] = sums[g] / fmaxf(cnts[g], 1.0f);
            mean_of_means += gm[g];
        }
        mean_of_means *= (1.0f / NGROUPS);

        float var = 0.0f;
#pragma unroll
        for (int g = 0; g < NGROUPS; ++g) {
            const float dv = gm[g] - mean_of_means;
            var += dv * dv;
        }
        var *= (1.0f / NGROUPS);

        out[0] = base_loss + 0.1f * sqrtf(var + 1e-8f);
    }
}

// ---------------------------------------------------------------------------
extern "C" void kernel_launch(void* const* d_in, const int* in_sizes, int n_in,
                              void* d_out, int out_size, void* d_ws, size_t ws_size,
                              hipStream_t stream) {
    const float* logits    = (const float*)d_in[0];
    const int*   targets   = (const int*)d_in[1];
    const int*   group_ids = (const int*)d_in[2];
    float*       out       = (float*)d_out;
    float*       partials  = (float*)d_ws;   // 1024 * 16 floats = 64 KB

    fairness_stage1<<<S1_BLOCKS, S1_THREADS, 0, stream>>>(logits, targets, group_ids, partials);
    fairness_stage2<<<1, 128, 0, stream>>>(partials, out);
}